// Model_1477468750330
// MI455X (gfx1250) — compile-verified
//
#include <hip/hip_runtime.h>
#include <hip/hip_bf16.h>
#include <math.h>

typedef __attribute__((ext_vector_type(2))) float v2f;
typedef __attribute__((ext_vector_type(8))) float v8f;

// ---------------------------------------------------------------------------
// Init: zero mask words, exclusive prefix-sum of m -> off, zero accumulators
// ---------------------------------------------------------------------------
__global__ void k_init(unsigned int* maskWords, int nMaskWords,
                       const int* m, int* off, int N, float* accum) {
    int tid = blockIdx.x * blockDim.x + threadIdx.x;
    if (tid < nMaskWords) maskWords[tid] = 0u;
    if (tid < N) {
        int s = 0;
        for (int k = 0; k < tid; ++k) s += m[k];
        off[tid] = s;
    }
    if (tid == 0) { accum[0] = 0.0f; accum[1] = 0.0f; }
}

// ---------------------------------------------------------------------------
// Scatter: mask[i][rel[e]] = 1 for edges e of sequence i
// ---------------------------------------------------------------------------
__global__ void k_scatter(const int* rel, const int* m, const int* off,
                          unsigned char* mask, int NREL) {
    int i  = blockIdx.x;
    int mi = m[i];
    int o  = off[i];
    for (int j = threadIdx.x; j < mi; j += blockDim.x) {
        int r = rel[o + j];
        mask[(size_t)i * NREL + r] = 1;
    }
}

// ---------------------------------------------------------------------------
// Gather-reduce: h[i] = sum_t emb[seq[i][t]]   (bandwidth-dominant phase)
// One block per sequence; token ids staged through LDS; float4 (b128) loads.
// ---------------------------------------------------------------------------
__global__ void k_embed_sum(const int* __restrict__ seq,
                            const float* __restrict__ emb,
                            float* __restrict__ h, int L, int NINP) {
    int i = blockIdx.x;
    int chunks = NINP >> 2;          // float4 chunks per row
    int c = threadIdx.x;
    extern __shared__ int stok[];
    for (int t = threadIdx.x; t < L; t += blockDim.x) stok[t] = seq[(size_t)i * L + t];
    __syncthreads();
    if (c >= chunks) return;
    float4 acc = make_float4(0.f, 0.f, 0.f, 0.f);
    const float4* emb4 = (const float4*)emb;
    for (int t = 0; t < L; ++t) {
        int tok = stok[t];
        float4 v = emb4[(size_t)tok * chunks + c];
        acc.x += v.x; acc.y += v.y; acc.z += v.z; acc.w += v.w;
    }
    ((float4*)h)[(size_t)i * chunks + c] = acc;
}

// ---------------------------------------------------------------------------
// WMMA GEMM (fp32, exact): logit = h @ W^T + b, fused logsigmoid/acc epilogue
// One wave per 16x16 output tile. K stepped by 4 via V_WMMA_F32_16X16X4_F32.
// OOB lanes clamp their row pointer to row 0: corrupted D rows/columns are
// local to that row/column and skipped by the guarded epilogue.
// ---------------------------------------------------------------------------
__global__ __launch_bounds__(32) void k_gemm_logp(
    const float* __restrict__ h, const float* __restrict__ W,
    const float* __restrict__ bias, const unsigned char* __restrict__ mask,
    float* accum, int N, int K, int NREL) {

    int mTile = blockIdx.x;
    int nTile = blockIdx.y;
    int lane  = threadIdx.x;
    int half  = lane >> 4;           // 0: this lane holds K=0,1  1: K=2,3
    int l16   = lane & 15;
    int kOff  = half * 2;

    int mRowA = mTile * 16 + l16;    // A-fragment row for this lane
    int nColB = nTile * 16 + l16;    // B-fragment column for this lane
    const float* hrow = h + (size_t)(mRowA < N    ? mRowA : 0) * K + kOff;
    const float* wrow = W + (size_t)(nColB < NREL ? nColB : 0) * K + kOff;

    v8f c = {};
    #pragma unroll 4
    for (int k0 = 0; k0 < K; k0 += 4) {
        v2f a = *(const v2f*)(hrow + k0);   // one global_load_b64
        v2f b = *(const v2f*)(wrow + k0);   // one global_load_b64
        // 8 args: (neg_a, A, neg_b, B, c_mod, C, reuse_a, reuse_b)
        c = __builtin_amdgcn_wmma_f32_16x16x4_f32(false, a, false, b,
                                                  (short)0, c, false, false);
    }

    // Epilogue: D layout -> lane l16 = N column; VGPR v -> M = v + 8*half
    int n = nTile * 16 + l16;
    float bn = (n < NREL) ? bias[n] : 0.0f;
    float lsum = 0.0f, asum = 0.0f;
    if (n < NREL) {
        #pragma unroll
        for (int v = 0; v < 8; ++v) {
            int mRow = mTile * 16 + v + 8 * half;
            if (mRow >= N) continue;
            float logit = c[v] + bn;
            bool mk = mask[(size_t)mRow * NREL + n] != 0;
            float term;
            if (mk) {
                // log_sigmoid(x) = min(x,0) - log1p(exp(-|x|))
                term = fminf(logit, 0.0f) - log1pf(expf(-fabsf(logit)));
            } else {
                float s = 1.0f / (1.0f + expf(-logit));
                term = logf(1.0f + 1e-5f - s);
            }
            lsum += term;
            asum += ((logit > 0.5f) == mk) ? 1.0f : 0.0f;
        }
    }
    // wave32 reduction
    #pragma unroll
    for (int o = 16; o > 0; o >>= 1) {
        lsum += __shfl_xor(lsum, o, 32);
        asum += __shfl_xor(asum, o, 32);
    }
    if (lane == 0) {
        atomicAdd(&accum[0], lsum);
        atomicAdd(&accum[1], asum);
    }
}

// ---------------------------------------------------------------------------
// Final normalize
// ---------------------------------------------------------------------------
__global__ void k_final(const float* accum, float* out, float invN, float invTot) {
    out[0] = accum[0] * invN;
    out[1] = accum[1] * invTot;
}

// ---------------------------------------------------------------------------
extern "C" void kernel_launch(void* const* d_in, const int* in_sizes, int n_in,
                              void* d_out, int out_size, void* d_ws, size_t ws_size,
                              hipStream_t stream) {
    const int*   seq = (const int*)d_in[0];
    const int*   m   = (const int*)d_in[1];
    const int*   rel = (const int*)d_in[2];
    const float* emb = (const float*)d_in[3];
    const float* W   = (const float*)d_in[4];
    const float* b   = (const float*)d_in[5];
    float* out = (float*)d_out;

    const int N    = in_sizes[1];            // m is (N,)
    const int L    = in_sizes[0] / N;        // seq is (N, L)
    const int NREL = in_sizes[5];            // b is (NREL,)
    const int NINP = in_sizes[4] / NREL;     // W is (NREL, NINP)

    // workspace layout
    float* h = (float*)d_ws;
    size_t hBytes = (size_t)N * NINP * sizeof(float);
    unsigned char* mask = (unsigned char*)d_ws + hBytes;
    size_t maskBytes = (((size_t)N * NREL) + 3) & ~(size_t)3;
    int*   off   = (int*)(mask + maskBytes);
    float* accum = (float*)(off + N);

    const int nMaskWords = (int)(maskBytes / 4);
    int initThreads = nMaskWords > N ? nMaskWords : N;
    k_init<<<(initThreads + 255) / 256, 256, 0, stream>>>(
        (unsigned int*)mask, nMaskWords, m, off, N, accum);

    k_scatter<<<N, 64, 0, stream>>>(rel, m, off, mask, NREL);

    int chunks = NINP / 4;                   // 128 threads for NINP=512
    k_embed_sum<<<N, chunks, L * (int)sizeof(int), stream>>>(seq, emb, h, L, NINP);

    dim3 gg((N + 15) / 16, (NREL + 15) / 16);
    k_gemm_logp<<<gg, 32, 0, stream>>>(h, W, b, mask, accum, N, NINP, NREL);

    k_final<<<1, 1, 0, stream>>>(accum, out, 1.0f / (float)N,
                                 1.0f / ((float)N * (float)NREL));
}